// ParBeamFPLayer_70248485093461
// MI455X (gfx1250) — compile-verified
//
#include <hip/hip_runtime.h>

// Parallel-beam forward projection (Radon) for MI455X / gfx1250.
// - TDM (tensor_load_to_lds) stages 33x512 f32 row-blocks into LDS,
//   double-buffered with TENSORcnt (s_wait_tensorcnt) + workgroup barriers.
// - All bilinear gathers come from LDS (ds_load_b32).
// - TDM pad feature inserts 1 DWORD per 256 DWORDs -> LDS row stride 514,
//   which spreads same-column gathers (theta ~ 90 deg) across banks.
// - __cosf/__sinf: theta in [0, pi), hardware v_cos/v_sin suffice; avoids the
//   Payne-Hanek SALU slow path.

#define IMG_N      512
#define RBLK       32
#define NBLOCKS    (IMG_N / RBLK)        // 16
#define ROWS_ST    (RBLK + 1)            // 33 rows staged (bilinear needs +1 row)
#define ROW_STRIDE 514                   // 512 data dwords + 2 pad dwords per row
#define BUF_DW     (ROWS_ST * ROW_STRIDE)

typedef unsigned int v4u __attribute__((ext_vector_type(4)));
typedef int          v8i __attribute__((ext_vector_type(8)));
typedef int          v4i __attribute__((ext_vector_type(4)));

// Build a TDM D# for a 2D tile: `rows` x 512 f32, global row stride 512,
// pad_enable with interval=256 DWORDs, amount=1 DWORD (table encoding 7 / 0).
__device__ __forceinline__ void tdm_load_rows(const float* gsrc,
                                              unsigned lds_byte,
                                              unsigned rows)
{
    unsigned long long ga = (unsigned long long)(uintptr_t)gsrc;
    v4u g0;
    g0[0] = 1u;                                            // count=1, user mode
    g0[1] = lds_byte;                                      // lds_addr (bytes)
    g0[2] = (unsigned)ga;                                  // global_addr[31:0]
    g0[3] = (unsigned)((ga >> 32) & 0x01FFFFFFull)         // global_addr[56:32]
          | (2u << 30);                                    // type=2 ("image")
    v8i g1;
    g1[0] = (int)((2u << 16)                               // data_size = 4 bytes
                | (1u << 20)                               // pad_enable
                | (7u << 22));                             // pad_interval: 256 DW
                                                           // pad_amount field 0 -> 1 DW
    g1[1] = (int)(((unsigned)IMG_N & 0xFFFFu) << 16);      // tensor_dim0 lo16
    g1[2] = (int)((((unsigned)IMG_N >> 16) & 0xFFFFu)      // tensor_dim0 hi16
                | ((rows & 0xFFFFu) << 16));               // tensor_dim1 lo16
    g1[3] = (int)(((rows >> 16) & 0xFFFFu)                 // tensor_dim1 hi16
                | ((unsigned)IMG_N << 16));                // tile_dim0 = 512
    g1[4] = (int)(rows & 0xFFFFu);                         // tile_dim1, tile_dim2=0
    g1[5] = IMG_N;                                         // tensor_dim0_stride lo32
    g1[6] = 0;                                             // stride hi / dim1_stride
    g1[7] = 0;
    v4i gz4 = {0, 0, 0, 0};                                // 2D tensor: groups 2/3 unused
    v8i gz8 = {0, 0, 0, 0, 0, 0, 0, 0};
    __builtin_amdgcn_tensor_load_to_lds(g0, g1, gz4, gz4, gz8, 0);
}

// Accumulate all t-steps of bin `s` whose sample-y falls in [lowY, highY)
// (half-open float partition across blocks => each step counted exactly once).
__device__ __forceinline__ float radon_block(const float* buf, float s,
                                             float ct, float st, float c,
                                             int y0, float lowY, float highY)
{
    const float ybase = s * st + c;   // y(j) = ybase + (j - c)*ct
    const float xbase = s * ct + c;   // x(j) = xbase - (j - c)*st
    int jlo, jhi;
    if (fabsf(ct) > 1e-7f) {
        float ja  = (lowY  - ybase) / ct + c;
        float jb  = (highY - ybase) / ct + c;
        float jmn = fminf(ja, jb), jmx = fmaxf(ja, jb);
        jlo = (int)fminf(fmaxf(jmn - 1.0f, 0.0f), (float)IMG_N);
        jhi = (int)fminf(fmaxf(jmx + 2.0f, 0.0f), (float)IMG_N);
    } else {  // theta ~ 90 deg: y nearly constant across j
        bool hit = (ybase >= lowY - 1.0f) && (ybase < highY + 1.0f);
        jlo = 0;
        jhi = hit ? IMG_N : 0;
    }

    float acc = 0.0f;
    for (int j = jlo; j < jhi; ++j) {
        float t = (float)j - c;
        float y = fmaf(t, ct, ybase);
        if (y >= lowY && y < highY) {
            float x  = fmaf(-t, st, xbase);
            float xf = floorf(x), yf = floorf(y);
            float wx = x - xf,    wy = y - yf;
            int xi = (int)xf, yi = (int)yf;
            bool vx0 = (xi >= 0)  && (xi < IMG_N);
            bool vx1 = (xi >= -1) && (xi < IMG_N - 1);
            bool vy0 = (yi >= 0);                // yi <= N-1 guaranteed by y < highY
            bool vy1 = ((yi + 1) < IMG_N);
            int xc0 = min(max(xi, 0), IMG_N - 1);
            int xc1 = min(max(xi + 1, 0), IMG_N - 1);
            int r0  = max(yi - y0, 0);           // yi == -1 only in block 0
            int r1  = min(yi - y0 + 1, ROWS_ST - 1);
            int p0  = xc0 + (xc0 >> 8);          // TDM pad: +1 for cols >= 256
            int p1  = xc1 + (xc1 >> 8);
            float f00 = buf[r0 * ROW_STRIDE + p0];
            float f01 = buf[r0 * ROW_STRIDE + p1];
            float f10 = buf[r1 * ROW_STRIDE + p0];
            float f11 = buf[r1 * ROW_STRIDE + p1];
            f00 = (vx0 && vy0) ? f00 : 0.0f;
            f01 = (vx1 && vy0) ? f01 : 0.0f;
            f10 = (vx0 && vy1) ? f10 : 0.0f;
            f11 = (vx1 && vy1) ? f11 : 0.0f;
            float top = fmaf(wx, f01 - f00, f00);
            float bot = fmaf(wx, f11 - f10, f10);
            acc = fmaf(wy, bot - top, acc + top);
        }
    }
    return acc;
}

extern "C" __global__ __launch_bounds__(256)
void radon_fp_kernel(const float* __restrict__ img, float* __restrict__ out)
{
    extern __shared__ float smem[];   // 2 * BUF_DW floats (double buffer)
    const int tid = threadIdx.x;
    const int b   = blockIdx.x >> 9;              // nViews = 512
    const int v   = blockIdx.x & (IMG_N - 1);

    const float theta = (float)v * 6.1359231515425649e-3f;  // pi / 512
    const float ct = __cosf(theta);   // theta in [0, pi): hw v_cos path is fine
    const float st = __sinf(theta);
    const float c  = (IMG_N - 1) * 0.5f;

    const float s0 = (float)tid - c;
    const float s1 = (float)(tid + 256) - c;
    float acc0 = 0.0f, acc1 = 0.0f;

    const float* imgb = img + (size_t)b * IMG_N * IMG_N;

    // Prefetch block 0 (wave 0 only; TDM is a wave-level op, TENSORcnt per-wave).
    if (tid < 32) {
        tdm_load_rows(imgb, (unsigned)(uintptr_t)&smem[0], ROWS_ST);
    }

    for (int k = 0; k < NBLOCKS; ++k) {
        if (tid < 32) {
            if (k + 1 < NBLOCKS) {
                const int y0n = (k + 1) * RBLK;
                const unsigned rows = (unsigned)min(ROWS_ST, IMG_N - y0n);
                tdm_load_rows(imgb + (size_t)y0n * IMG_N,
                              (unsigned)(uintptr_t)&smem[((k + 1) & 1) * BUF_DW],
                              rows);
                __builtin_amdgcn_s_wait_tensorcnt(1);  // block k load done (in-order)
            } else {
                __builtin_amdgcn_s_wait_tensorcnt(0);
            }
        }
        __syncthreads();   // block k's tile visible to all waves

        const float* buf  = &smem[(k & 1) * BUF_DW];
        const int   y0    = k * RBLK;
        const float lowY  = (k == 0) ? -1.0f : (float)y0;
        const float highY = (k == NBLOCKS - 1) ? (float)IMG_N
                                               : (float)(y0 + RBLK);

        acc0 += radon_block(buf, s0, ct, st, c, y0, lowY, highY);
        acc1 += radon_block(buf, s1, ct, st, c, y0, lowY, highY);

        __syncthreads();   // all reads of buf done before TDM overwrites it
    }

    const size_t o = ((size_t)b * IMG_N + (size_t)v) * IMG_N;
    out[o + tid]       = acc0;
    out[o + tid + 256] = acc1;
}

extern "C" void kernel_launch(void* const* d_in, const int* in_sizes, int n_in,
                              void* d_out, int out_size, void* d_ws, size_t ws_size,
                              hipStream_t stream)
{
    const float* img = (const float*)d_in[0];
    float*       out = (float*)d_out;
    const int batch  = in_sizes[0] / (IMG_N * IMG_N);   // 4

    const dim3 grid((unsigned)(batch * IMG_N));
    const dim3 block(256);
    const size_t shmem = 2u * BUF_DW * sizeof(float);   // ~133 KB of 320 KB WGP LDS

    radon_fp_kernel<<<grid, block, shmem, stream>>>(img, out);
}